// LSTM_SOFT_ATT_57990648430683
// MI455X (gfx1250) — compile-verified
//
#include <hip/hip_runtime.h>
#include <hip/hip_bf16.h>
#include <math.h>

// ---------------------------------------------------------------------------
// LSTM + Bahdanau attention + log-softmax projection for MI455X (gfx1250).
// GEMMs on V_WMMA_F32_16X16X4_F32 (exact fp32); att staging via
// global_load_async_to_lds_b128 (ASYNCcnt) on the dominant einsum.
// B=256, A=196, R=1024, I=1024, V=10000.
// ---------------------------------------------------------------------------

typedef __attribute__((ext_vector_type(2))) float v2f;
typedef __attribute__((ext_vector_type(8))) float v8f;

constexpr int kB = 256;
constexpr int kA = 196;
constexpr int kR = 1024;
constexpr int kV = 10000;
constexpr int kH = 4096;   // 4*R gate rows

__device__ __forceinline__ float sigmoidf_(float x) {
    return 1.0f / (1.0f + expf(-x));
}

__device__ __forceinline__ void wait_async0() {
#if __has_builtin(__builtin_amdgcn_s_wait_asynccnt)
    __builtin_amdgcn_s_wait_asynccnt(0);
#else
    asm volatile("s_wait_asynccnt 0x0" ::: "memory");
#endif
}

// Branch-free K-loop: C += A(16xK from pa, masked) * B(Kx16 from pb).
// pa/pb are per-lane base pointers already offset by row and kb=2*(lane/16).
// amask==1.0f (constant) folds away in the full-tile path.
__device__ __forceinline__ v8f wmma_kloop(const float* __restrict__ pa,
                                          const float* __restrict__ pb,
                                          int K, float amask, v8f c) {
#pragma unroll 8
    for (int k = 0; k < K; k += 4) {
        v2f a, b;
        a.x = pa[k] * amask;
        a.y = pa[k + 1] * amask;
        b.x = pb[k];
        b.y = pb[k + 1];
        c = __builtin_amdgcn_wmma_f32_16x16x4_f32(false, a, false, b,
                                                  (short)0, c, false, false);
    }
    return c;
}

// C[16,16] += Wa[m0:m0+16, 0:K] * Wb[n0:n0+16, 0:K]^T, Wa rows zero-padded
// past Mmax. Layouts per CDNA5 ISA 7.12.2:
//   A 16x4 f32: M = lane%16, VGPR0/1 = K + 2*(lane/16) + {0,1}
//   B  4x16 f32: N = lane%16, same K selection
//   C 16x16 f32: VGPR v -> M = v + 8*(lane/16), N = lane%16
__device__ __forceinline__ v8f wmma_gemm_f32(const float* __restrict__ Wa, int m0,
                                             int Mmax, int lda,
                                             const float* __restrict__ Wb, int n0,
                                             int ldb, int K, v8f c) {
    const int lane = threadIdx.x & 31;
    const int mr = m0 + (lane & 15);
    const int kb = 2 * (lane >> 4);
    const int mrc = (mr < Mmax) ? mr : (Mmax - 1);   // clamp, no branchy loads
    const float* __restrict__ pa = Wa + (size_t)mrc * lda + kb;
    const float* __restrict__ pb = Wb + (size_t)(n0 + (lane & 15)) * ldb + kb;
    if (m0 + 16 <= Mmax)                              // block-uniform fast path
        return wmma_kloop(pa, pb, K, 1.0f, c);
    return wmma_kloop(pa, pb, K, (mr < Mmax) ? 1.0f : 0.0f, c);
}

// --------------------------- K0: att_h = prev_h @ h2a_w^T + h2a_b ----------
__global__ void k_att_h(const float* __restrict__ prev_h,
                        const float* __restrict__ h2a_w,
                        const float* __restrict__ h2a_b,
                        float* __restrict__ att_h) {
    const int m0 = blockIdx.x * 16;   // over A=196 (13 tiles, last partial)
    const int n0 = blockIdx.y * 16;   // over B=256
    v8f c = {};
    c = wmma_gemm_f32(h2a_w, m0, kA, kR, prev_h, n0, kR, kR, c);
    const int lane = threadIdx.x & 31;
    const int n = n0 + (lane & 15);
#pragma unroll
    for (int v = 0; v < 8; ++v) {
        const int m = m0 + v + 8 * (lane >> 4);
        if (m < kA) att_h[(size_t)n * kA + m] = c[v] + h2a_b[m];
    }
}

// --------------------------- K1: attention scores --------------------------
// score[p] = sum_a d2d_w[a]*tanh(att_h[p] + a2a_b[a] + sum_r a2a_w[a,r]*att[p,r])
// p = b*A + i flattened (B*A = 50176 = 3136 tiles of 16).
// 4 waves/block; 16 att rows staged through LDS in 16KB K-chunks via
// global_load_async_to_lds_b128; B-fragments then read from LDS (ds_load_b64).
constexpr int kKC = 256;   // K chunk
__global__ void k_score(const float* __restrict__ att,
                        const float* __restrict__ a2a_w,
                        const float* __restrict__ a2a_b,
                        const float* __restrict__ att_h,
                        const float* __restrict__ d2d_w,
                        const float* __restrict__ d2d_b,
                        float* __restrict__ score) {
    __shared__ float satt[16 * kKC];     // 16 KB staging for 16 att rows
    __shared__ float s_red[4][16];

    const int p0 = blockIdx.x * 16;
    const int tid = threadIdx.x;
    const int lane = tid & 31;
    const int wave = tid >> 5;           // 0..3

    v8f c[4];
#pragma unroll
    for (int j = 0; j < 4; ++j) c[j] = (v8f){};

    const int kb = 2 * (lane >> 4);

    for (int kc = 0; kc < kR; kc += kKC) {
        __syncthreads();   // WAR: previous chunk's LDS reads done
        // async DMA: 16 rows x kKC floats, 16B per lane per issue (ASYNCcnt)
        for (int idx = tid; idx < 16 * (kKC / 4); idx += 128) {
            const int r = idx / (kKC / 4);
            const int cc = (idx % (kKC / 4)) * 4;
            const float* g = att + (size_t)(p0 + r) * kR + kc + cc;
            const unsigned lds_addr =
                (unsigned)(uintptr_t)(satt + r * kKC + cc);
            asm volatile("global_load_async_to_lds_b128 %0, %1, off"
                         :: "v"(lds_addr), "v"(g)
                         : "memory");
        }
        wait_async0();
        __syncthreads();
#pragma unroll
        for (int j = 0; j < 4; ++j) {
            const int at = wave + 4 * j;     // uniform per wave
            if (at < 13) {
                const int mr = at * 16 + (lane & 15);
                const int mrc = (mr < kA) ? mr : (kA - 1);
                const float* __restrict__ pa =
                    a2a_w + (size_t)mrc * kR + kc + kb;
                const float* __restrict__ pb = satt + (lane & 15) * kKC + kb;
                if (at * 16 + 16 <= kA)      // tiles 0..11: mask-free
                    c[j] = wmma_kloop(pa, pb, kKC, 1.0f, c[j]);
                else                         // tile 12: rows 196..207 zeroed
                    c[j] = wmma_kloop(pa, pb, kKC,
                                      (mr < kA) ? 1.0f : 0.0f, c[j]);
            }
        }
    }

    // epilogue: bias + tanh + d2d dot, reduce over a within lane, then waves
    const float ah = att_h[p0 + (lane & 15)];
    float partial = 0.0f;
#pragma unroll
    for (int j = 0; j < 4; ++j) {
        const int at = wave + 4 * j;
        if (at < 13) {
#pragma unroll
            for (int v = 0; v < 8; ++v) {
                const int m = at * 16 + v + 8 * (lane >> 4);
                const float w  = (m < kA) ? d2d_w[m] : 0.0f;
                const float bb = (m < kA) ? a2a_b[m] : 0.0f;
                partial += w * tanhf(c[j][v] + bb + ah);
            }
        }
    }
    partial += __shfl_xor(partial, 16);   // fold M-halves (lane <-> lane+16)
    if (lane < 16) s_red[wave][lane] = partial;
    __syncthreads();
    if (tid < 16)
        score[p0 + tid] =
            s_red[0][tid] + s_red[1][tid] + s_red[2][tid] + s_red[3][tid] +
            d2d_b[0];
}

// --------------------------- K2: softmax over i (A=196) --------------------
__global__ void k_softmax(const float* __restrict__ score,
                          float* __restrict__ weight) {
    const int b = blockIdx.x;
    const int tid = threadIdx.x;   // 256 threads
    __shared__ float red[256];
    const float v = (tid < kA) ? score[(size_t)b * kA + tid] : -INFINITY;
    red[tid] = v;
    __syncthreads();
    for (int s = 128; s > 0; s >>= 1) {
        if (tid < s) red[tid] = fmaxf(red[tid], red[tid + s]);
        __syncthreads();
    }
    const float mx = red[0];
    __syncthreads();
    const float e = (tid < kA) ? expf(v - mx) : 0.0f;
    red[tid] = e;
    __syncthreads();
    for (int s = 128; s > 0; s >>= 1) {
        if (tid < s) red[tid] += red[tid + s];
        __syncthreads();
    }
    if (tid < kA) weight[(size_t)b * kA + tid] = e / red[0];
}

// --------------------------- K3: att_res[b,r] = sum_i att[b,i,r]*w[b,i] ----
__global__ void k_attres(const float* __restrict__ att,
                         const float* __restrict__ weight,
                         float* __restrict__ att_res) {
    const int b = blockIdx.x;
    const int tid = threadIdx.x;   // 256 threads
    __shared__ float w_s[kA];
    if (tid < kA) w_s[tid] = weight[(size_t)b * kA + tid];
    __syncthreads();
    const int r = blockIdx.y * 256 + tid;
    const float* __restrict__ base = att + (size_t)b * kA * kR + r;
    float acc = 0.0f;
    for (int i = 0; i < kA; ++i) {
        if (i + 8 < kA) __builtin_prefetch(base + (size_t)(i + 8) * kR, 0, 1);
        acc += base[(size_t)i * kR] * w_s[i];
    }
    att_res[(size_t)b * kR + r] = acc;
}

// --------------------------- K4: fused 3-GEMM gate pre-activations ---------
__global__ void k_gates_gemm(const float* __restrict__ x,
                             const float* __restrict__ prev_h,
                             const float* __restrict__ att_res,
                             const float* __restrict__ i2h_w,
                             const float* __restrict__ h2h_w,
                             const float* __restrict__ r2a_w,
                             const float* __restrict__ i2h_b,
                             const float* __restrict__ h2h_b,
                             const float* __restrict__ r2a_b,
                             float* __restrict__ sums) {
    const int m0 = blockIdx.x * 16;   // over 4096 gate rows
    const int n0 = blockIdx.y * 16;   // over B=256
    v8f c = {};
    c = wmma_gemm_f32(i2h_w, m0, kH, kR, x,       n0, kR, kR, c);
    c = wmma_gemm_f32(h2h_w, m0, kH, kR, prev_h,  n0, kR, kR, c);
    c = wmma_gemm_f32(r2a_w, m0, kH, kR, att_res, n0, kR, kR, c);
    const int lane = threadIdx.x & 31;
    const int n = n0 + (lane & 15);
#pragma unroll
    for (int v = 0; v < 8; ++v) {
        const int m = m0 + v + 8 * (lane >> 4);
        sums[(size_t)n * kH + m] = c[v] + i2h_b[m] + h2h_b[m] + r2a_b[m];
    }
}

// --------------------------- K5: LSTM cell elementwise ---------------------
__global__ void k_cell(const float* __restrict__ sums,
                       const float* __restrict__ prev_c,
                       float* __restrict__ next_c,
                       float* __restrict__ next_h) {
    const int idx = blockIdx.x * 256 + threadIdx.x;  // B*R
    const int b = idx >> 10;
    const int r = idx & (kR - 1);
    const float* __restrict__ s = sums + (size_t)b * kH;
    const float ig = sigmoidf_(s[r]);
    const float fg = sigmoidf_(s[kR + r]);
    const float og = sigmoidf_(s[2 * kR + r]);
    const float gt = tanhf(s[3 * kR + r]);
    const float nc = fg * prev_c[idx] + ig * gt;
    next_c[idx] = nc;
    next_h[idx] = og * tanhf(nc);
}

// --------------------------- K6: logits = next_h @ proj_w^T + proj_b -------
__global__ void k_proj(const float* __restrict__ next_h,
                       const float* __restrict__ proj_w,
                       const float* __restrict__ proj_b,
                       float* __restrict__ logits) {
    const int m0 = blockIdx.x * 16;   // over V=10000 (625 exact tiles)
    const int n0 = blockIdx.y * 16;   // over B=256
    v8f c = {};
    c = wmma_gemm_f32(proj_w, m0, kV, kR, next_h, n0, kR, kR, c);
    const int lane = threadIdx.x & 31;
    const int n = n0 + (lane & 15);
#pragma unroll
    for (int v = 0; v < 8; ++v) {
        const int m = m0 + v + 8 * (lane >> 4);
        logits[(size_t)n * kV + m] = c[v] + proj_b[m];
    }
}

// --------------------------- K7: in-place log-softmax over V ---------------
__global__ void k_logsoftmax(float* __restrict__ logits) {
    const int b = blockIdx.x;
    const int tid = threadIdx.x;   // 256 threads
    __shared__ float red[256];
    float* __restrict__ row = logits + (size_t)b * kV;
    float mx = -INFINITY;
    for (int v = tid; v < kV; v += 256) mx = fmaxf(mx, row[v]);
    red[tid] = mx;
    __syncthreads();
    for (int s = 128; s > 0; s >>= 1) {
        if (tid < s) red[tid] = fmaxf(red[tid], red[tid + s]);
        __syncthreads();
    }
    mx = red[0];
    __syncthreads();
    float sum = 0.0f;
    for (int v = tid; v < kV; v += 256) sum += expf(row[v] - mx);
    red[tid] = sum;
    __syncthreads();
    for (int s = 128; s > 0; s >>= 1) {
        if (tid < s) red[tid] += red[tid + s];
        __syncthreads();
    }
    const float lse = mx + logf(red[0]);
    for (int v = tid; v < kV; v += 256) row[v] -= lse;
}

// ---------------------------------------------------------------------------
extern "C" void kernel_launch(void* const* d_in, const int* in_sizes, int n_in,
                              void* d_out, int out_size, void* d_ws,
                              size_t ws_size, hipStream_t stream) {
    (void)in_sizes; (void)n_in; (void)out_size; (void)ws_size;

    const float* x      = (const float*)d_in[0];
    const float* att    = (const float*)d_in[1];
    const float* prev_c = (const float*)d_in[2];
    const float* prev_h = (const float*)d_in[3];
    const float* a2a_w  = (const float*)d_in[4];
    const float* a2a_b  = (const float*)d_in[5];
    const float* h2a_w  = (const float*)d_in[6];
    const float* h2a_b  = (const float*)d_in[7];
    const float* d2d_w  = (const float*)d_in[8];
    const float* d2d_b  = (const float*)d_in[9];
    const float* i2h_w  = (const float*)d_in[10];
    const float* i2h_b  = (const float*)d_in[11];
    const float* h2h_w  = (const float*)d_in[12];
    const float* h2h_b  = (const float*)d_in[13];
    const float* r2a_w  = (const float*)d_in[14];
    const float* r2a_b  = (const float*)d_in[15];
    const float* proj_w = (const float*)d_in[16];
    const float* proj_b = (const float*)d_in[17];

    float* out_c  = (float*)d_out;                 // [B,R]
    float* out_h  = out_c + (size_t)kB * kR;       // [B,R]
    float* out_ls = out_h + (size_t)kB * kR;       // [B,V]

    // workspace layout (floats): ~5.9 MB total
    float* ws      = (float*)d_ws;
    float* att_h   = ws;                           // B*A
    float* score   = att_h + (size_t)kB * kA;      // B*A
    float* weight  = score + (size_t)kB * kA;      // B*A
    float* att_res = weight + (size_t)kB * kA;     // B*R
    float* sums    = att_res + (size_t)kB * kR;    // B*4R

    k_att_h<<<dim3((kA + 15) / 16, kB / 16), 32, 0, stream>>>(
        prev_h, h2a_w, h2a_b, att_h);
    k_score<<<dim3(kB * kA / 16), 128, 0, stream>>>(
        att, a2a_w, a2a_b, att_h, d2d_w, d2d_b, score);
    k_softmax<<<kB, 256, 0, stream>>>(score, weight);
    k_attres<<<dim3(kB, kR / 256), 256, 0, stream>>>(att, weight, att_res);
    k_gates_gemm<<<dim3(kH / 16, kB / 16), 32, 0, stream>>>(
        x, prev_h, att_res, i2h_w, h2h_w, r2a_w, i2h_b, h2h_b, r2a_b, sums);
    k_cell<<<dim3(kB * kR / 256), 256, 0, stream>>>(sums, prev_c, out_c, out_h);
    k_proj<<<dim3(kV / 16, kB / 16), 32, 0, stream>>>(
        out_h, proj_w, proj_b, out_ls);
    k_logsoftmax<<<kB, 256, 0, stream>>>(out_ls);
}